// DiscreteRotation_1305670058160
// MI455X (gfx1250) — compile-verified
//
#include <hip/hip_runtime.h>

#define B_ 64
#define H_ 224
#define W_ 224
#define C_ 32
#define CHUNKS 32
#define POS_PER_CHUNK ((H_ * W_) / CHUNKS) // 1568

typedef __attribute__((ext_vector_type(2))) float v2f;
typedef __attribute__((ext_vector_type(8))) float v8f;

// ---------------------------------------------------------------------------
// Kernel 1: deterministic partial spatial sums.
// Grid: B*CHUNKS blocks, 256 threads. Each thread owns 4 channels (float4)
// and strides over 32-position groups; LDS tree reduce; plain stores (no
// atomics -> bitwise deterministic across graph replays).
// partial layout: [B][CHUNKS][C] floats.
// ---------------------------------------------------------------------------
__global__ void dr_partial_sum(const float* __restrict__ x,
                               float* __restrict__ partial) {
    const int blk   = blockIdx.x;
    const int b     = blk >> 5;      // /CHUNKS
    const int chunk = blk & 31;
    const int t  = threadIdx.x;      // 0..255
    const int cg = t & 7;            // channel group (4 channels)
    const int p  = t >> 3;           // 0..31 position lane

    const float4* __restrict__ x4 = (const float4*)x;
    // float4 index of (b, pos, cg) = (b*H*W + pos)*8 + cg
    size_t base = ((size_t)b * (size_t)(H_ * W_) +
                   (size_t)chunk * (size_t)POS_PER_CHUNK) * 8 + (size_t)cg;

    float4 s = make_float4(0.f, 0.f, 0.f, 0.f);
    #pragma unroll 7
    for (int it = 0; it < POS_PER_CHUNK / 32; ++it) {
        float4 v = x4[base + (size_t)(p + it * 32) * 8];
        s.x += v.x; s.y += v.y; s.z += v.z; s.w += v.w;
    }

    __shared__ float4 sd[256];
    sd[t] = s;
    __syncthreads();
    // strides are multiples of 8 -> same channel group folds onto itself
    for (int st = 128; st >= 8; st >>= 1) {
        if (t < st) {
            float4 a = sd[t], c = sd[t + st];
            a.x += c.x; a.y += c.y; a.z += c.z; a.w += c.w;
            sd[t] = a;
        }
        __syncthreads();
    }
    if (t < 8) {
        float4* p4 = (float4*)partial;
        p4[(size_t)(b * CHUNKS + chunk) * 8 + t] = sd[t];
    }
}

// ---------------------------------------------------------------------------
// Kernel 2: means -> logits (V_WMMA_F32_16X16X4_F32) -> argmax -> k[b].
// 1 block, 128 threads = 4 waves. Wave w owns batch rows 16w..16w+15 (M),
// N = 4 real columns padded to 16 with zeros, K = 32 accumulated as 8
// chained k=4 fp32 WMMAs. Bias pre-seeded into the C accumulator.
// A layout (ISA 16x4 f32): lane l -> M = l&15, K pair base = (l>>4)*2.
// C/D layout: VGPR r, lane l -> M = r + 8*(l>>4), N = l&15.
// ---------------------------------------------------------------------------
__global__ void dr_logits_argmax(const float* __restrict__ partial,
                                 const float* __restrict__ Wc,   // [C][4]
                                 const float* __restrict__ bias, // [4]
                                 int* __restrict__ kbuf) {
    __shared__ float mean_s[B_ * C_];    // 8 KB
    __shared__ float logit_s[B_ * 16];   // 4 KB

    const int t = threadIdx.x; // 0..127
    const float invHW = 1.0f / (float)(H_ * W_);

    // Deterministic reduction of CHUNKS partials per (b, c)
    for (int e = t; e < B_ * C_; e += 128) {
        const int b = e >> 5;
        const int c = e & 31;
        float s = 0.f;
        #pragma unroll
        for (int ch = 0; ch < CHUNKS; ++ch)
            s += partial[(size_t)(b * CHUNKS + ch) * C_ + c];
        mean_s[e] = s * invHW;
    }
    __syncthreads();

    const int w    = t >> 5;   // wave 0..3
    const int l    = t & 31;   // lane
    const int mrow = l & 15;   // M within tile (A frag rows / D column group)
    const int hi   = l >> 4;   // lane-half selects K pair / M+8 in D
    const int n    = l & 15;   // N for B/C/D fragments

    const float bn = (n < 4) ? bias[n] : 0.0f;
    v8f acc;
    #pragma unroll
    for (int r = 0; r < 8; ++r) acc[r] = bn;

    const int gb = w * 16 + mrow; // global batch row for A fragment
    #pragma unroll
    for (int kstep = 0; kstep < 8; ++kstep) {
        const int k0 = kstep * 4 + hi * 2;
        v2f a, bb;
        a.x = mean_s[gb * C_ + k0];
        a.y = mean_s[gb * C_ + k0 + 1];
        bb.x = (n < 4) ? Wc[k0 * 4 + n]       : 0.0f;
        bb.y = (n < 4) ? Wc[(k0 + 1) * 4 + n] : 0.0f;
        acc = __builtin_amdgcn_wmma_f32_16x16x4_f32(
            /*neg_a=*/false, a, /*neg_b=*/false, bb,
            /*c_mod=*/(short)0, acc, /*reuse_a=*/false, /*reuse_b=*/false);
    }

    // Scatter D tile to LDS: row M = r + 8*hi within this wave's 16-row tile
    #pragma unroll
    for (int r = 0; r < 8; ++r) {
        const int gm = w * 16 + r + 8 * hi;
        logit_s[gm * 16 + n] = acc[r];
    }
    __syncthreads();

    if (t < B_) {
        float best = logit_s[t * 16 + 0];
        int   bi   = 0;
        #pragma unroll
        for (int nn = 1; nn < 4; ++nn) {
            const float v = logit_s[t * 16 + nn];
            if (v > best) { best = v; bi = nn; } // strict > : first-max, jnp.argmax
        }
        kbuf[t] = bi;
    }
}

// ---------------------------------------------------------------------------
// Kernel 3: per-sample rot90^k permutation copy, float4 granularity.
// Grid: B*H blocks (one output row each), 256 threads: 32 positions x 8
// channel-groups per iteration, 7 iterations. Output fully coalesced;
// input reads are whole 128B lines regardless of rotation.
// jnp.rot90 CCW: k1: out[i,j]=x[j,W-1-i]; k2: out[i,j]=x[H-1-i,W-1-j];
// k3: out[i,j]=x[H-1-j,i].
// ---------------------------------------------------------------------------
__global__ void dr_rotate(const float* __restrict__ x,
                          const int* __restrict__ kbuf,
                          float* __restrict__ out) {
    const int blk = blockIdx.x;        // 0 .. B*H-1
    const int b   = blk / H_;
    const int i   = blk - b * H_;
    const int kk  = kbuf[b];

    const int t  = threadIdx.x;
    const int cg = t & 7;
    const int p  = t >> 3;

    const float4* __restrict__ x4 = (const float4*)x;
    float4* __restrict__       o4 = (float4*)out;

    const size_t obase = (size_t)(b * H_ + i) * (size_t)W_ * 8;
    const size_t ibase = (size_t)b * (size_t)(H_ * W_) * 8;

    #pragma unroll 7
    for (int j = p; j < W_; j += 32) {
        int si, sj;
        switch (kk) {
            case 1:  si = j;          sj = W_ - 1 - i; break;
            case 2:  si = H_ - 1 - i; sj = W_ - 1 - j; break;
            case 3:  si = H_ - 1 - j; sj = i;          break;
            default: si = i;          sj = j;          break;
        }
        o4[obase + (size_t)j * 8 + cg] =
            x4[ibase + ((size_t)si * W_ + sj) * 8 + cg];
    }
}

// ---------------------------------------------------------------------------
extern "C" void kernel_launch(void* const* d_in, const int* in_sizes, int n_in,
                              void* d_out, int out_size, void* d_ws, size_t ws_size,
                              hipStream_t stream) {
    (void)in_sizes; (void)n_in; (void)out_size; (void)ws_size;

    const float* x    = (const float*)d_in[0]; // [B,H,W,C]
    const float* Wc   = (const float*)d_in[1]; // [C,4]
    const float* bias = (const float*)d_in[2]; // [4]
    float*       out  = (float*)d_out;

    // Workspace: partial sums [B][CHUNKS][C] floats, then k[B] ints.
    float* partial = (float*)d_ws;
    int*   kbuf    = (int*)((char*)d_ws + (size_t)B_ * CHUNKS * C_ * sizeof(float));

    dr_partial_sum <<<B_ * CHUNKS, 256, 0, stream>>>(x, partial);
    dr_logits_argmax<<<1, 128, 0, stream>>>(partial, Wc, bias, kbuf);
    dr_rotate      <<<B_ * H_, 256, 0, stream>>>(x, kbuf, out);
}